// DNA_65360812310552
// MI455X (gfx1250) — compile-verified
//
#include <hip/hip_runtime.h>

#define NN 10000
#define NE 160000
#define CC 128
#define HH 8
#define LL 5
#define NF 14
#define DO 16
#define ETOT (NE + NN)          // edges + self loops = 170000
#define XST (6 * CC)            // floats per node in x_all (6 layer slots)

typedef __attribute__((ext_vector_type(2))) float v2f;
typedef __attribute__((ext_vector_type(4))) float v4f;
typedef __attribute__((ext_vector_type(8))) float v8f;

__device__ __forceinline__ v8f wmma4(v2f a, v2f b, v8f c) {
    // D(16x16,f32) = A(16x4,f32) * B(4x16,f32) + C
    return __builtin_amdgcn_wmma_f32_16x16x4_f32(false, a, false, b, (short)0, c,
                                                 false, false);
}

__device__ __forceinline__ void lds_fence() {
    asm volatile("s_wait_dscnt 0" ::: "memory");
}

// sum over the 16 lanes of each wave half (xor butterfly masks 1,2,4,8)
__device__ __forceinline__ float red16(float t) {
#if __has_builtin(__builtin_amdgcn_permlane16)
    union { float f; unsigned u; } a, b;
    a.f = t;
    b.u = __builtin_amdgcn_permlane16(a.u, a.u, 0x67452301u, 0xEFCDAB89u, true, false); // ^1
    a.f += b.f;
    b.u = __builtin_amdgcn_permlane16(a.u, a.u, 0x54761032u, 0xDCFE98BAu, true, false); // ^2
    a.f += b.f;
    b.u = __builtin_amdgcn_permlane16(a.u, a.u, 0x32107654u, 0xBA98FEDCu, true, false); // ^4
    a.f += b.f;
    b.u = __builtin_amdgcn_permlane16(a.u, a.u, 0xFEDCBA98u, 0x76543210u, true, false); // ^8
    a.f += b.f;
    return a.f;
#else
    t += __shfl_xor(t, 1, 32);
    t += __shfl_xor(t, 2, 32);
    t += __shfl_xor(t, 4, 32);
    t += __shfl_xor(t, 8, 32);
    return t;
#endif
}

// blockdiag weight value for pair p: B[K][N] (K=input, N=output within pair)
__device__ __forceinline__ float bdiag(const float* W, int p, int K, int n) {
    if (K < 8 && n < 8)   return W[(2 * p) * 64 + K * 8 + n];
    if (K >= 8 && n >= 8) return W[(2 * p + 1) * 64 + (K - 8) * 8 + (n - 8)];
    return 0.0f;
}

// ---- init: deg=1 (self loop), zero x_all layer slots 1..5 ----
__global__ void init_kernel(float* deg, float* xall) {
    size_t stride = (size_t)gridDim.x * blockDim.x;
    size_t total = (size_t)NN * 5 * CC;
    for (size_t i = blockIdx.x * (size_t)blockDim.x + threadIdx.x; i < total; i += stride) {
        size_t node = i / (5 * CC);
        size_t r = i % (5 * CC);
        xall[node * XST + CC + r] = 0.0f;
        if (i < NN) deg[i] = 1.0f;
    }
}

__global__ void deg_kernel(const int* ei, float* deg) {
    int e = blockIdx.x * blockDim.x + threadIdx.x;
    if (e < NE) atomicAdd(&deg[ei[NE + e]], 1.0f);   // col = edge_index[1]
}

__global__ void dis_kernel(float* deg) {
    int i = blockIdx.x * blockDim.x + threadIdx.x;
    if (i < NN) deg[i] = rsqrtf(deg[i]);             // in-place -> dis
}

// ---- lin1: h = relu(x @ W1 + b1) into x_all slot 0 ----
__global__ void lin1_kernel(const float* __restrict__ x, const float* __restrict__ W,
                            const float* __restrict__ b, float* __restrict__ xall) {
    __shared__ float wl[NF * CC];
    __shared__ float bl[CC];
    int tid = threadIdx.x;   // blockDim = 128, tid == output channel
    for (int i = tid; i < NF * CC; i += blockDim.x) wl[i] = W[i];
    if (tid < CC) bl[tid] = b[tid];
    __syncthreads();
    for (int node = blockIdx.x; node < NN; node += gridDim.x) {
        const float* xr = x + (size_t)node * NF;
        float acc = bl[tid];
        #pragma unroll
        for (int k = 0; k < NF; k++) acc += xr[k] * wl[k * CC + tid];
        xall[(size_t)node * XST + tid] = fmaxf(acc, 0.0f);
    }
}

// ---- per-layer edge attention + scatter (WMMA grouped linears) ----
// block = 128 threads = 4 waves; each wave owns a 16-edge tile.
__global__ void edge_attn_kernel(const int* __restrict__ ei, const float* __restrict__ dis,
                                 float* __restrict__ xall,
                                 const float* __restrict__ Wq, const float* __restrict__ bq,
                                 const float* __restrict__ Wk, const float* __restrict__ bk,
                                 const float* __restrict__ Wv, const float* __restrict__ bv,
                                 int layer) {
    __shared__ float wfQ[8 * 4 * 32 * 2];   // [p][kk][lane][2] expanded B fragments
    __shared__ float wfK[8 * 4 * 32 * 2];
    __shared__ float wfV[8 * 4 * 32 * 2];
    __shared__ float bQ[CC], bK[CC], bV[CC];
    __shared__ float attnS[4][8 * 5 * 16];  // per-wave scores->attn, layout [p][j][m]

    int tid = threadIdx.x;
    for (int i = tid; i < 1024; i += blockDim.x) {
        int ln = i & 31, kk = (i >> 5) & 3, p = i >> 7;
        int n = ln & 15, h2 = ln >> 4;
        int K0 = kk * 4 + h2 * 2;
        wfQ[i * 2 + 0] = bdiag(Wq, p, K0, n);     wfQ[i * 2 + 1] = bdiag(Wq, p, K0 + 1, n);
        wfK[i * 2 + 0] = bdiag(Wk, p, K0, n);     wfK[i * 2 + 1] = bdiag(Wk, p, K0 + 1, n);
        wfV[i * 2 + 0] = bdiag(Wv, p, K0, n);     wfV[i * 2 + 1] = bdiag(Wv, p, K0 + 1, n);
    }
    if (tid < CC) { bQ[tid] = bq[tid]; bK[tid] = bk[tid]; bV[tid] = bv[tid]; }
    __syncthreads();

    int lane = tid & 31;
    int wib = tid >> 5;
    int gwave = blockIdx.x * 4 + wib;
    int nwaves = gridDim.x * 4;
    int ntiles = (ETOT + 15) / 16;
    float* myA = attnS[wib];
    int h2 = lane >> 4;
    int mi = lane & 15;

    for (int tile = gwave; tile < ntiles; tile += nwaves) {
        int rm = 0, cm = 0;
        float nm = 0.0f;
        int et = tile * 16 + lane;
        if (lane < 16) {
            if (et < NE)        { rm = ei[et]; cm = ei[NE + et]; nm = dis[rm] * dis[cm]; }
            else if (et < ETOT) { rm = cm = et - NE;             nm = dis[rm] * dis[cm]; }
        }
        int rM = __shfl(rm, mi, 32);   // node for my A-fragment row M = lane%16
        int cM = __shfl(cm, mi, 32);

        // ---- pass 1: q,k fragments -> scores ----
        for (int p = 0; p < 8; p++) {
            const float* qrow = xall + (size_t)cM * XST + layer * CC + p * 16;
            float bqv = bQ[p * 16 + mi];
            v8f qf;
            #pragma unroll
            for (int v = 0; v < 8; v++) qf[v] = bqv;
            #pragma unroll
            for (int kk = 0; kk < 4; kk++) {
                v2f a = *(const v2f*)(qrow + kk * 4 + 2 * h2);
                v2f bb = *(const v2f*)(wfQ + ((p * 4 + kk) * 32 + lane) * 2);
                qf = wmma4(a, bb, qf);
            }
            for (int j = 0; j <= layer; j++) {
                const float* krow = xall + (size_t)rM * XST + j * CC + p * 16;
                float bkv = bK[p * 16 + mi];
                v8f kf;
                #pragma unroll
                for (int v = 0; v < 8; v++) kf[v] = bkv;
                #pragma unroll
                for (int kk = 0; kk < 4; kk++) {
                    v2f a = *(const v2f*)(krow + kk * 4 + 2 * h2);
                    v2f bb = *(const v2f*)(wfK + ((p * 4 + kk) * 32 + lane) * 2);
                    kf = wmma4(a, bb, kf);
                }
                #pragma unroll
                for (int v = 0; v < 8; v++) {
                    float t = red16(qf[v] * kf[v]);      // reduce head dim over 16 lanes
                    if (mi == 0) {
                        int M = v + 8 * h2;
                        myA[(p * 5 + j) * 16 + M] = t * 0.25f;   // * 1/sqrt(CH)
                    }
                }
            }
        }
        lds_fence();

        // ---- softmax over layers, per (m,h) ----
        int lj = layer + 1;
        for (int it = 0; it < 4; it++) {
            int idx = it * 32 + lane;                 // 128 (m,h) pairs
            int m = idx & 15, h = idx >> 4;
            float* s = myA + h * 5 * 16 + m;          // stride 16 over j
            float mx = -1e30f;
            for (int j = 0; j < lj; j++) mx = fmaxf(mx, s[j * 16]);
            float e[5]; float sum = 0.0f;
            for (int j = 0; j < lj; j++) { e[j] = __expf(s[j * 16] - mx); sum += e[j]; }
            float inv = 1.0f / sum;
            for (int j = 0; j < lj; j++) s[j * 16] = e[j] * inv;
        }
        lds_fence();

        // ---- pass 2: v fragments, weighted sum, scatter-add ----
        for (int p = 0; p < 8; p++) {
            v8f out;
            #pragma unroll
            for (int v = 0; v < 8; v++) out[v] = 0.0f;
            for (int j = 0; j <= layer; j++) {
                const float* vrow = xall + (size_t)rM * XST + j * CC + p * 16;
                float bvv = bV[p * 16 + mi];
                v8f vf;
                #pragma unroll
                for (int v = 0; v < 8; v++) vf[v] = bvv;
                #pragma unroll
                for (int kk = 0; kk < 4; kk++) {
                    v2f a = *(const v2f*)(vrow + kk * 4 + 2 * h2);
                    v2f bb = *(const v2f*)(wfV + ((p * 4 + kk) * 32 + lane) * 2);
                    vf = wmma4(a, bb, vf);
                }
                // attn weights for this lane's 8 rows (m = v + 8*h2), contiguous
                const float* ab = myA + (p * 5 + j) * 16 + 8 * h2;
                v4f a0 = *(const v4f*)(ab);
                v4f a1 = *(const v4f*)(ab + 4);
                #pragma unroll
                for (int v = 0; v < 4; v++) out[v]     += a0[v] * vf[v];
                #pragma unroll
                for (int v = 0; v < 4; v++) out[v + 4] += a1[v] * vf[v + 4];
            }
            #pragma unroll
            for (int v = 0; v < 8; v++) {
                int M = v + 8 * h2;
                float nrm = __shfl(nm, M, 32);
                int cn = __shfl(cm, M, 32);
                atomicAdd(xall + (size_t)cn * XST + (layer + 1) * CC + p * 16 + mi,
                          nrm * out[v]);
            }
        }
    }
}

__global__ void relu_kernel(float* xall, int layer) {
    int idx = blockIdx.x * blockDim.x + threadIdx.x;
    if (idx < NN * CC) {
        int node = idx >> 7, c = idx & 127;
        float* p = xall + (size_t)node * XST + (layer + 1) * CC + c;
        *p = fmaxf(*p, 0.0f);
    }
}

__global__ void lin2_kernel(const float* __restrict__ xall, const float* __restrict__ W,
                            const float* __restrict__ b, float* __restrict__ out) {
    __shared__ float wl[CC * DO];
    int tid = threadIdx.x;
    for (int i = tid; i < CC * DO; i += blockDim.x) wl[i] = W[i];
    __syncthreads();
    int total = NN * DO;
    for (int idx = blockIdx.x * blockDim.x + tid; idx < total; idx += gridDim.x * blockDim.x) {
        int node = idx >> 4, o = idx & 15;
        const float* xr = xall + (size_t)node * XST + 5 * CC;
        float acc = b[o];
        #pragma unroll 8
        for (int k = 0; k < CC; k++) acc += xr[k] * wl[k * DO + o];
        out[idx] = acc;
    }
}

extern "C" void kernel_launch(void* const* d_in, const int* in_sizes, int n_in,
                              void* d_out, int out_size, void* d_ws, size_t ws_size,
                              hipStream_t stream) {
    const float* x      = (const float*)d_in[0];
    const int*   ei     = (const int*)d_in[1];
    const float* lin1_w = (const float*)d_in[2];
    const float* lin1_b = (const float*)d_in[3];
    const float* Wq     = (const float*)d_in[4];
    const float* bq     = (const float*)d_in[5];
    const float* Wk     = (const float*)d_in[6];
    const float* bk     = (const float*)d_in[7];
    const float* Wv     = (const float*)d_in[8];
    const float* bv     = (const float*)d_in[9];
    const float* lin2_w = (const float*)d_in[10];
    const float* lin2_b = (const float*)d_in[11];
    float* out = (float*)d_out;

    char* ws = (char*)d_ws;
    float* deg  = (float*)ws;                 // N floats (becomes dis in-place)
    float* xall = (float*)(ws + 65536);       // N * 6 * 128 floats (~30.7 MB)

    init_kernel<<<1024, 256, 0, stream>>>(deg, xall);
    deg_kernel<<<(NE + 255) / 256, 256, 0, stream>>>(ei, deg);
    dis_kernel<<<(NN + 255) / 256, 256, 0, stream>>>(deg);
    lin1_kernel<<<1024, 128, 0, stream>>>(x, lin1_w, lin1_b, xall);
    for (int l = 0; l < LL; l++) {
        edge_attn_kernel<<<1024, 128, 0, stream>>>(ei, deg, xall,
                                                   Wq + l * 1024, bq + l * CC,
                                                   Wk + l * 1024, bk + l * CC,
                                                   Wv + l * 1024, bv + l * CC, l);
        relu_kernel<<<(NN * CC + 255) / 256, 256, 0, stream>>>(xall, l);
    }
    lin2_kernel<<<512, 256, 0, stream>>>(xall, lin2_w, lin2_b, out);
}